// GatLinear_44727789421066
// MI455X (gfx1250) — compile-verified
//
#include <hip/hip_runtime.h>
#include <hip/hip_bf16.h>
#include <math.h>

typedef __attribute__((ext_vector_type(2))) float v2f;
typedef __attribute__((ext_vector_type(4))) float v4f;
typedef __attribute__((ext_vector_type(8))) float v8f;

#define B_ 64
#define N_ 2048
#define D_ 512
#define NT_ 8                      // n-chunks for attn_sum partials
#define ROWS_PER_BLOCK 128         // 8 waves x 16 rows
#define PB2 (N_ / ROWS_PER_BLOCK)  // 16 alpha-blocks per batch

__device__ __forceinline__ float wave_max32(float v) {
  #pragma unroll
  for (int off = 16; off > 0; off >>= 1)
    v = fmaxf(v, __shfl_xor(v, off, 32));
  return v;
}

// ---------------------------------------------------------------------------
// K1: qdot[b] = Q[b,:].Wq + bias   via V_WMMA_F32_16X16X4_F32
// One block, 4 waves; wave w handles Q rows [16w, 16w+16).
// A (16x4 f32): lane L -> row L&15, K-pair 2*(L>>4); per-lane contiguous float2.
// B (4x16 f32): Wk chunk in column 0 only (lanes 0 and 16 hold k={0,1}/{2,3}).
// C column 0: lane 0 -> rows 0..7 (c[0..7]), lane 16 -> rows 8..15.
// ---------------------------------------------------------------------------
__global__ void qdot_kernel(const float* __restrict__ Q,
                            const float* __restrict__ W,
                            const float* __restrict__ bias,
                            float* __restrict__ qdot) {
  const int lane = threadIdx.x & 31;
  const int wave = threadIdx.x >> 5;  // 0..3
  const int hi   = lane >> 4;
  const int mrow = lane & 15;
  const int row0 = wave * 16;
  const float* rp = Q + (size_t)(row0 + mrow) * D_ + 2 * hi;
  const float* Wq = W;  // first D entries of W[0,:]
  v8f c = {};
  for (int d0 = 0; d0 < D_; d0 += 4) {
    v2f a = *(const v2f*)(rp + d0);
    v4f wk = *(const v4f*)(Wq + d0);
    v2f bb;
    bb.x = (lane == 0) ? wk.x : ((lane == 16) ? wk.z : 0.0f);
    bb.y = (lane == 0) ? wk.y : ((lane == 16) ? wk.w : 0.0f);
    c = __builtin_amdgcn_wmma_f32_16x16x4_f32(false, a, false, bb,
                                              (short)0, c, false, false);
  }
  const float bv = bias[0];
  if (lane == 0 || lane == 16) {
    const int rbase = row0 + ((lane == 16) ? 8 : 0);
    #pragma unroll
    for (int r = 0; r < 8; ++r) qdot[rbase + r] = c[r] + bv;
  }
}

// ---------------------------------------------------------------------------
// K2: alpha[b,n] = qdot[b] + K[b,n,:].Wk, masked; streams K (256 MB).
// 1024 blocks x 256 thr (8 waves x 16 rows). Writes alpha into the weight
// output region; per-block max -> pmax for the softmax.
// ---------------------------------------------------------------------------
__global__ void alpha_kernel(const float* __restrict__ K,
                             const float* __restrict__ W,
                             const int* __restrict__ adj,
                             const float* __restrict__ qdot,
                             float* __restrict__ out_alpha,
                             float* __restrict__ pmax) {
  __shared__ float smax[8];
  const int blk  = blockIdx.x;
  const int b    = blk >> 4;   // / PB2
  const int tile = blk & 15;
  const int lane = threadIdx.x & 31;
  const int wave = threadIdx.x >> 5;  // 0..7
  const int hi   = lane >> 4;
  const int mrow = lane & 15;
  const int n0   = tile * ROWS_PER_BLOCK + wave * 16;
  const float* rp = K + ((size_t)b * N_ + n0 + mrow) * D_ + 2 * hi;
  const float* Wk = W + D_;
  v8f c = {};
  for (int d0 = 0; d0 < D_; d0 += 4) {
    v2f a = *(const v2f*)(rp + d0);
    v4f wk = *(const v4f*)(Wk + d0);
    v2f bb;
    bb.x = (lane == 0) ? wk.x : ((lane == 16) ? wk.z : 0.0f);
    bb.y = (lane == 0) ? wk.y : ((lane == 16) ? wk.w : 0.0f);
    c = __builtin_amdgcn_wmma_f32_16x16x4_f32(false, a, false, bb,
                                              (short)0, c, false, false);
  }
  float lmax = -3.4e38f;
  const float qd = qdot[b];
  if (lane == 0 || lane == 16) {
    const int rbase = n0 + ((lane == 16) ? 8 : 0);
    #pragma unroll
    for (int r = 0; r < 8; ++r) {
      const int n = rbase + r;
      float al = c[r] + qd;
      al -= (1.0f - (float)adj[(size_t)b * N_ + n]) * 1e30f;
      out_alpha[(size_t)b * N_ + n] = al;
      lmax = fmaxf(lmax, al);
    }
  }
  lmax = wave_max32(lmax);
  if (lane == 0) smax[wave] = lmax;
  __syncthreads();
  if (threadIdx.x == 0) {
    float m = smax[0];
    #pragma unroll
    for (int i = 1; i < 8; ++i) m = fmaxf(m, smax[i]);
    pmax[blk] = m;
  }
}

// ---------------------------------------------------------------------------
// K3: per batch: max over pmax partials, then sum of exp(alpha - max).
// ---------------------------------------------------------------------------
__global__ void softmax_stats_kernel(const float* __restrict__ alpha,
                                     const float* __restrict__ pmax,
                                     float* __restrict__ maxv,
                                     float* __restrict__ sumv) {
  __shared__ float sred[256];
  __shared__ float smx;
  const int b = blockIdx.x;
  const int t = threadIdx.x;
  float m = -3.4e38f;
  if (t < PB2) m = pmax[b * PB2 + t];
  sred[t] = m;
  __syncthreads();
  for (int s = 128; s > 0; s >>= 1) {
    if (t < s) sred[t] = fmaxf(sred[t], sred[t + s]);
    __syncthreads();
  }
  if (t == 0) smx = sred[0];
  __syncthreads();
  const float mx = smx;
  float local = 0.0f;
  for (int n = t; n < N_; n += 256)
    local += expf(alpha[(size_t)b * N_ + n] - mx);
  __syncthreads();
  sred[t] = local;
  __syncthreads();
  for (int s = 128; s > 0; s >>= 1) {
    if (t < s) sred[t] += sred[t + s];
    __syncthreads();
  }
  if (t == 0) { maxv[b] = mx; sumv[b] = sred[0]; }
}

// ---------------------------------------------------------------------------
// K4: normalize weights (write final attn_weight) and accumulate partial
// attn_sum = sum_n w[n]*V[n,:] over a 256-row chunk; streams V (256 MB).
// 512 blocks x 256 thr; thread t owns columns {2t, 2t+1}; deterministic.
// ---------------------------------------------------------------------------
__global__ void attn_kernel(const float* __restrict__ V,
                            const float* __restrict__ maxv,
                            const float* __restrict__ sumv,
                            float* __restrict__ wout,
                            float* __restrict__ partial) {
  __shared__ float sw[256];
  const int blk = blockIdx.x;
  const int b   = blk / NT_;
  const int nt  = blk % NT_;
  const int n0  = nt * (N_ / NT_);  // 256-row chunk
  const int t   = threadIdx.x;
  const float mx  = maxv[b];
  const float inv = 1.0f / sumv[b];
  {
    const int n = n0 + t;
    const float al = wout[(size_t)b * N_ + n];
    const float w = expf(al - mx) * inv;
    wout[(size_t)b * N_ + n] = w;   // final attn_weight
    sw[t] = w;
  }
  __syncthreads();
  const v2f* Vp = (const v2f*)(V + ((size_t)b * N_ + n0) * D_);
  v2f acc; acc.x = 0.0f; acc.y = 0.0f;
  for (int n = 0; n < (N_ / NT_); n += 4) {
    if (n + 24 < (N_ / NT_))
      __builtin_prefetch((const void*)(Vp + (size_t)(n + 24) * (D_ / 2) + t), 0, 0);
    #pragma unroll
    for (int j = 0; j < 4; ++j) {
      const float w = sw[n + j];
      const v2f v = Vp[(size_t)(n + j) * (D_ / 2) + t];
      acc.x += w * v.x;
      acc.y += w * v.y;
    }
  }
  *(v2f*)(partial + ((size_t)b * NT_ + nt) * D_ + 2 * t) = acc;
}

// ---------------------------------------------------------------------------
// K5: deterministic reduction of the NT_ partials -> attn_sum output.
// ---------------------------------------------------------------------------
__global__ void reduce_attn_kernel(const float* __restrict__ partial,
                                   float* __restrict__ out2) {
  const int idx = blockIdx.x * 256 + threadIdx.x;  // 0 .. B*D
  const int b = idx / D_;
  const int d = idx % D_;
  float s = 0.0f;
  #pragma unroll
  for (int nt = 0; nt < NT_; ++nt)
    s += partial[((size_t)b * NT_ + nt) * D_ + d];
  out2[idx] = s;
}

extern "C" void kernel_launch(void* const* d_in, const int* in_sizes, int n_in,
                              void* d_out, int out_size, void* d_ws, size_t ws_size,
                              hipStream_t stream) {
  (void)in_sizes; (void)n_in; (void)out_size; (void)ws_size;
  const float* Q    = (const float*)d_in[0];
  const float* K    = (const float*)d_in[1];
  const float* V    = (const float*)d_in[2];
  const int*   adj  = (const int*)d_in[3];
  const float* W    = (const float*)d_in[4];
  const float* bias = (const float*)d_in[5];

  float* out      = (float*)d_out;
  float* w_out    = out;                          // B*N attn_weight
  float* attn_out = out + (size_t)B_ * N_;        // B*D attn_sum

  float* ws      = (float*)d_ws;
  float* qdot    = ws;                  // 64
  float* pmax    = ws + 64;             // 1024
  float* maxv    = ws + 64 + 1024;      // 64
  float* sumv    = maxv + 64;           // 64
  float* partial = ws + 1216;           // B*NT_*D = 262144 floats (1 MB)

  qdot_kernel<<<1, 128, 0, stream>>>(Q, W, bias, qdot);
  alpha_kernel<<<B_ * PB2, 256, 0, stream>>>(K, W, adj, qdot, w_out, pmax);
  softmax_stats_kernel<<<B_, 256, 0, stream>>>(w_out, pmax, maxv, sumv);
  attn_kernel<<<B_ * NT_, 256, 0, stream>>>(V, maxv, sumv, w_out, partial);
  reduce_attn_kernel<<<(B_ * D_) / 256, 256, 0, stream>>>(partial, attn_out);
}